// GaussianKernel_14851996909997
// MI455X (gfx1250) — compile-verified
//
#include <hip/hip_runtime.h>
#include <cmath>

typedef __attribute__((ext_vector_type(2))) float v2f;
typedef __attribute__((ext_vector_type(4))) float v4f;
typedef __attribute__((ext_vector_type(8))) float v8f;

#define NSPLIT 8

__device__ __forceinline__ v2f ld2(const float* p) {
  return *reinterpret_cast<const v2f*>(p);
}

// ---------------------------------------------------------------------------
// Kernel 0: half squared norms of every row (test rows then train rows).
// ---------------------------------------------------------------------------
__global__ void kde_norms(const float* __restrict__ test,
                          const float* __restrict__ train,
                          float* __restrict__ norms,
                          int nTest, int nTrain) {
  int tid = blockIdx.x * blockDim.x + threadIdx.x;
  if (tid >= nTest + nTrain) return;
  const float* src = (tid < nTest) ? (test + (size_t)tid * 16)
                                   : (train + (size_t)(tid - nTest) * 16);
  float s = 0.0f;
#pragma unroll
  for (int i = 0; i < 4; ++i) {
    v4f v = *reinterpret_cast<const v4f*>(src + i * 4);
    s = fmaf(v.x, v.x, s);
    s = fmaf(v.y, v.y, s);
    s = fmaf(v.z, v.z, s);
    s = fmaf(v.w, v.w, s);
  }
  norms[tid] = 0.5f * s;
}

// ---------------------------------------------------------------------------
// Kernel 1: one wave per (16-row tile, column split).
// GEMM tile via V_WMMA_F32_16X16X4_F32 (K=4 x 4 = d=16), fused online
// logsumexp over the train axis in registers.
// ---------------------------------------------------------------------------
__global__ __launch_bounds__(32)
void kde_main(const float* __restrict__ test,
              const float* __restrict__ train,
              const float* __restrict__ ht,     // 0.5*||test_m||^2
              const float* __restrict__ hr,     // 0.5*||train_n||^2
              float* __restrict__ partials,     // [nTest][NSPLIT][2] = (max, sum)
              int tilesPerSplit, int nColTiles) {
  const int lane = threadIdx.x;
  const int l = lane & 15;
  const int h = lane >> 4;
  const int rowBase = blockIdx.x * 16;
  const int split = blockIdx.y;

  // A matrix (16x4 fp32 per chunk): lane (l,h) holds test[rowBase+l][4k+2h, 4k+2h+1]
  const float* arow = test + (size_t)(rowBase + l) * 16 + 2 * h;
  v2f a0 = ld2(arow), a1 = ld2(arow + 4), a2 = ld2(arow + 8), a3 = ld2(arow + 12);

  float negHT[8];
#pragma unroll
  for (int r = 0; r < 8; ++r) negHT[r] = -ht[rowBase + r + 8 * h];

  float mrun[8], srun[8];
#pragma unroll
  for (int r = 0; r < 8; ++r) { mrun[r] = -__builtin_inff(); srun[r] = 0.0f; }

  int t0 = split * tilesPerSplit;
  int t1 = t0 + tilesPerSplit;
  if (t1 > nColTiles) t1 = nColTiles;

  for (int t = t0; t < t1; ++t) {
    const int colBase = t * 16;
    // B matrix (4x16 fp32 per chunk): lane (l,h) holds train[colBase+l][4k+2h, +1]
    const float* brow = train + (size_t)(colBase + l) * 16 + 2 * h;
    v2f b0 = ld2(brow), b1 = ld2(brow + 4), b2 = ld2(brow + 8), b3 = ld2(brow + 12);
    float hR = hr[colBase + l];

    v8f acc = {};
    acc = __builtin_amdgcn_wmma_f32_16x16x4_f32(false, a0, false, b0, (short)0, acc, false, false);
    acc = __builtin_amdgcn_wmma_f32_16x16x4_f32(false, a1, false, b1, (short)0, acc, false, false);
    acc = __builtin_amdgcn_wmma_f32_16x16x4_f32(false, a2, false, b2, (short)0, acc, false, false);
    acc = __builtin_amdgcn_wmma_f32_16x16x4_f32(false, a3, false, b3, (short)0, acc, false, false);

    // acc[r] = dot(test[rowBase+r+8h], train[colBase+l])
    // x = -0.5*||t - s||^2 = dot - 0.5||t||^2 - 0.5||s||^2
#pragma unroll
    for (int r = 0; r < 8; ++r) {
      float x = acc[r] + negHT[r] - hR;
      float M = fmaxf(mrun[r], x);
      srun[r] = srun[r] * __expf(mrun[r] - M) + __expf(x - M);
      mrun[r] = M;
    }
  }

  // Merge the 16 lanes of each half (xor 1,2,4,8 stays within the half).
#pragma unroll
  for (int r = 0; r < 8; ++r) {
#pragma unroll
    for (int off = 1; off <= 8; off <<= 1) {
      float mo = __shfl_xor(mrun[r], off, 32);
      float so = __shfl_xor(srun[r], off, 32);
      float M = fmaxf(mrun[r], mo);
      srun[r] = srun[r] * __expf(mrun[r] - M) + so * __expf(mo - M);
      mrun[r] = M;
    }
  }

  if (l == 0) {
#pragma unroll
    for (int r = 0; r < 8; ++r) {
      int row = rowBase + r + 8 * h;
      float* p = partials + ((size_t)row * NSPLIT + split) * 2;
      p[0] = mrun[r];
      p[1] = srun[r];
    }
  }
}

// ---------------------------------------------------------------------------
// Kernel 2: merge NSPLIT partial (max,sum) pairs per test row, apply log + Z.
// ---------------------------------------------------------------------------
__global__ void kde_finalize(const float* __restrict__ partials,
                             float* __restrict__ out, int nTest, float Z) {
  int tid = blockIdx.x * blockDim.x + threadIdx.x;
  if (tid >= nTest) return;
  float M = -__builtin_inff(), S = 0.0f;
#pragma unroll
  for (int i = 0; i < NSPLIT; ++i) {
    float mi = partials[((size_t)tid * NSPLIT + i) * 2 + 0];
    float si = partials[((size_t)tid * NSPLIT + i) * 2 + 1];
    if (!(mi > -__builtin_inff())) continue;  // empty split
    float nM = fmaxf(M, mi);
    S = S * __expf(M - nM) + si * __expf(mi - nM);
    M = nM;
  }
  out[tid] = M + __logf(S) - Z;
}

// ---------------------------------------------------------------------------
extern "C" void kernel_launch(void* const* d_in, const int* in_sizes, int n_in,
                              void* d_out, int out_size, void* d_ws, size_t ws_size,
                              hipStream_t stream) {
  const float* test  = (const float*)d_in[0];
  const float* train = (const float*)d_in[1];
  float* out = (float*)d_out;

  const int nTest  = in_sizes[0] / 16;
  const int nTrain = in_sizes[1] / 16;

  float* ws = (float*)d_ws;
  float* norms = ws;                 // [nTest + nTrain] half squared norms
  float* ht = norms;
  float* hr = norms + nTest;
  float* partials = norms + nTest + nTrain;  // [nTest][NSPLIT][2]

  const int nColTiles = nTrain / 16;
  const int tilesPerSplit = (nColTiles + NSPLIT - 1) / NSPLIT;

  const int total = nTest + nTrain;
  kde_norms<<<(total + 255) / 256, 256, 0, stream>>>(test, train, norms, nTest, nTrain);

  dim3 grid(nTest / 16, NSPLIT);
  kde_main<<<grid, 32, 0, stream>>>(test, train, ht, hr, partials,
                                    tilesPerSplit, nColTiles);

  // Z = 0.5*d*log(2*pi) + d*log(h) + log(n), d=16, h=1
  const float Z = 0.5f * 16.0f * logf(2.0f * 3.14159265358979323846f)
                + logf((float)nTrain);
  kde_finalize<<<(nTest + 255) / 256, 256, 0, stream>>>(partials, out, nTest, Z);
}